// _DSTGCN_43198781063683
// MI455X (gfx1250) — compile-verified
//
#include <hip/hip_runtime.h>
#include <math.h>

// ---------------------------------------------------------------------------
// DSTGCN forward for gfx1250 (MI455X).
// Dense matmuls (FC + graph-conv feature transforms) run on V_WMMA_F32_16X16X4_F32
// (exact fp32 => matches reference numerics), templated on (FI,FO) so fragment
// loads are unconditional/branchless and the WMMA chain is fully unrolled.
// Edge aggregation uses global_atomic_add_f32 (float4-vectorized gather);
// BN stats use LDS ds_add_f32 reduction.
// ---------------------------------------------------------------------------

typedef float v2f __attribute__((ext_vector_type(2)));
typedef float v8f __attribute__((ext_vector_type(8)));

// ---------------------------------------------------------------------------
// out[rows,FO] = act( (rowscale? .* in[rows,FI]) @ W[FI,FO] + bias )
// One wave handles TWO 16-row tiles (sharing B fragments) x all column tiles.
// WMMA fp32 fragment layouts per CDNA5 ISA 7.12.2:
//   A(16x4):  lane = m + 16*(k/2), reg = k&1   (every lane holds row m = lane&15)
//   B(4x16):  lane = n + 16*(k/2), reg = k&1
//   C/D:      lanes 0-15: M=r, lanes 16-31: M=8+r; N = lane&15
// K-padding: A reads a clamped (in-bounds) index; the matching B entry is zero,
// so the product contributes nothing. Row tails: clamped reads, guarded stores.
// ---------------------------------------------------------------------------
template <int FI, int FO>
__global__ __launch_bounds__(128)
void k_wmma_dense_t(const float* __restrict__ in, int ldi,
                    const float* __restrict__ W,          // [FI, FO] row-major
                    const float* __restrict__ bias,       // nullable
                    const float* __restrict__ rowscale,   // nullable (on[] per node)
                    int rdiv,                              // node = row / rdiv
                    float* __restrict__ out, int ldo,
                    long rows, int relu)
{
    constexpr int KS = (FI + 3) / 4;      // k-steps of 4
    constexpr int CT = (FO + 15) / 16;    // column tiles (1 or 2)
    const int  lane = threadIdx.x & 31;
    const long wave = (long)blockIdx.x * (blockDim.x >> 5) + (threadIdx.x >> 5);
    const long nPairs = (rows + 31) / 32;
    if (wave >= nPairs) return;           // uniform per wave
    const int  lo = lane & 15;
    const int  hi = lane >> 4;
    const long r0 = wave * 32;

    // ---- B fragments (weights; shared by both row tiles), zero-padded via select
    v2f bf[CT][KS];
#pragma unroll
    for (int ct = 0; ct < CT; ++ct) {
        const int col  = ct * 16 + lo;
        const int colc = (col < FO) ? col : (FO - 1);
#pragma unroll
        for (int ks = 0; ks < KS; ++ks) {
            const int k0 = ks * 4 + hi * 2;
            const int ka = (k0 < FI) ? k0 : (FI - 1);
            const int kb = (k0 + 1 < FI) ? (k0 + 1) : (FI - 1);
            float b0 = W[(long)ka * FO + colc];
            float b1 = W[(long)kb * FO + colc];
            if (!(k0 < FI && col < FO))     b0 = 0.0f;
            if (!(k0 + 1 < FI && col < FO)) b1 = 0.0f;
            bf[ct][ks].x = b0;
            bf[ct][ks].y = b1;
        }
    }

    // ---- A fragments for two row tiles (row-scaled), all loads unconditional
    v2f af[2][KS];
#pragma unroll
    for (int mt = 0; mt < 2; ++mt) {
        long row = r0 + mt * 16 + lo;
        if (row > rows - 1) row = rows - 1;          // in-bounds clamp (tail)
        float scale = 1.0f;
        if (rowscale) scale = rowscale[row / (long)rdiv];
        const float* rp = in + row * (long)ldi;
#pragma unroll
        for (int ks = 0; ks < KS; ++ks) {
            const int k0 = ks * 4 + hi * 2;
            const int ka = (k0 < FI) ? k0 : (FI - 1);
            const int kb = (k0 + 1 < FI) ? (k0 + 1) : (FI - 1);
            af[mt][ks].x = rp[ka] * scale;
            af[mt][ks].y = rp[kb] * scale;
        }
    }

    // ---- WMMA chains (two independent accumulator chains per column tile)
    v8f acc[2][CT];
    const v8f vzero = {0.f, 0.f, 0.f, 0.f, 0.f, 0.f, 0.f, 0.f};
#pragma unroll
    for (int mt = 0; mt < 2; ++mt)
#pragma unroll
        for (int ct = 0; ct < CT; ++ct)
            acc[mt][ct] = vzero;

#pragma unroll
    for (int ks = 0; ks < KS; ++ks)
#pragma unroll
        for (int mt = 0; mt < 2; ++mt)
#pragma unroll
            for (int ct = 0; ct < CT; ++ct)
                acc[mt][ct] = __builtin_amdgcn_wmma_f32_16x16x4_f32(
                    /*neg_a=*/false, af[mt][ks], /*neg_b=*/false, bf[ct][ks],
                    /*c_mod=*/(short)0, acc[mt][ct],
                    /*reuse_a=*/false, /*reuse_b=*/false);

    // ---- epilogue: bias + relu + guarded stores
#pragma unroll
    for (int mt = 0; mt < 2; ++mt) {
#pragma unroll
        for (int ct = 0; ct < CT; ++ct) {
            const int n = ct * 16 + lo;
            if (n < FO) {
                const float bi = bias ? bias[n] : 0.0f;
#pragma unroll
                for (int r = 0; r < 8; ++r) {
                    const long m = r0 + mt * 16 + r + 8 * hi;
                    if (m < rows) {
                        float v = acc[mt][ct][r] + bi;
                        if (relu) v = fmaxf(v, 0.0f);
                        out[m * (long)ldo + n] = v;
                    }
                }
            }
        }
    }
}

// Scalar fallback (only for unexpected shape combos; never hit for this net).
__global__ void k_dense_fallback(const float* __restrict__ in, int ldi, int fi,
                                 const float* __restrict__ W,
                                 const float* __restrict__ bias,
                                 const float* __restrict__ rowscale, int rdiv,
                                 float* __restrict__ out, int ldo, int fo,
                                 long rows, int relu)
{
    long row = (long)blockIdx.x * blockDim.x + threadIdx.x;
    if (row >= rows) return;
    float sc = rowscale ? rowscale[row / (long)rdiv] : 1.0f;
    for (int o = 0; o < fo; ++o) {
        float v = 0.0f;
        for (int k = 0; k < fi; ++k) v += in[row * (long)ldi + k] * W[(long)k * fo + o];
        v *= sc;
        if (bias) v += bias[o];
        if (relu) v = fmaxf(v, 0.0f);
        out[row * (long)ldo + o] = v;
    }
}

// ---------------------------------------------------------------------------
__global__ void k_zero(float* __restrict__ p, long n)
{
    long i = (long)blockIdx.x * blockDim.x + threadIdx.x;
    if (i < n) p[i] = 0.0f;
}

__global__ void k_count(const int* __restrict__ idx, int* __restrict__ deg, long E)
{
    long i = (long)blockIdx.x * blockDim.x + threadIdx.x;
    if (i < E) atomicAdd(&deg[idx[i]], 1);
}

__global__ void k_rsqrtdeg(const int* __restrict__ deg, float* __restrict__ o, long n)
{
    long i = (long)blockIdx.x * blockDim.x + threadIdx.x;
    if (i < n) o[i] = rsqrtf(fmaxf((float)deg[i], 1.0f));
}

// agg[dst[e]*vpn + c] += val[src[e]*vpn + c]   (scalar path)
__global__ void k_scatter(const int* __restrict__ src, const int* __restrict__ dst,
                          const float* __restrict__ val, float* __restrict__ agg,
                          long E, int vpn)
{
    long i = (long)blockIdx.x * blockDim.x + threadIdx.x;
    if (i >= E * (long)vpn) return;
    long e = i / vpn;
    int  c = (int)(i - e * vpn);
    atomicAdd(&agg[(long)dst[e] * vpn + c], val[(long)src[e] * vpn + c]);
}

// float4 gather + 4 atomics per thread (vpn % 4 == 0)
__global__ void k_scatter4(const int* __restrict__ src, const int* __restrict__ dst,
                           const float* __restrict__ val, float* __restrict__ agg,
                           long E, int vpn)
{
    const int q = vpn >> 2;
    long i = (long)blockIdx.x * blockDim.x + threadIdx.x;
    if (i >= E * (long)q) return;
    long e = i / q;
    int  c = (int)(i - e * q) << 2;
    const float4 v = *(const float4*)(val + (long)src[e] * vpn + c);
    float* a = agg + (long)dst[e] * vpn + c;
    atomicAdd(a + 0, v.x);
    atomicAdd(a + 1, v.y);
    atomicAdd(a + 2, v.z);
    atomicAdd(a + 3, v.w);
}

// z = agg * inn[node] + bias; accumulate per-channel sum / sumsq into stats[0..2f)
__global__ void k_finish(const float* __restrict__ agg, const float* __restrict__ inn,
                         int rdiv, const float* __restrict__ bias,
                         float* __restrict__ z, float* __restrict__ stats,
                         long rows, int f)
{
    __shared__ float sst[64];                 // 2*f <= 40
    const int tid = threadIdx.x;
    if (tid < 2 * f) sst[tid] = 0.0f;
    __syncthreads();
    const long row = (long)blockIdx.x * blockDim.x + tid;
    if (row < rows) {
        const float innv = inn[row / (long)rdiv];
        for (int c = 0; c < f; ++c) {
            float v = agg[row * (long)f + c] * innv + bias[c];
            z[row * (long)f + c] = v;
            atomicAdd(&sst[c], v);            // ds_add_f32
            atomicAdd(&sst[f + c], v * v);
        }
    }
    __syncthreads();
    if (tid < 2 * f) atomicAdd(&stats[tid], sst[tid]);
}

// out = relu( g*(z-mu)*rsqrt(var+1e-5)+be ) [+ res]
__global__ void k_bn(const float* __restrict__ z, const float* __restrict__ stats,
                     const float* __restrict__ g, const float* __restrict__ be,
                     const float* __restrict__ res, float* __restrict__ out,
                     long rows, int f, float invcount)
{
    long row = (long)blockIdx.x * blockDim.x + threadIdx.x;
    if (row >= rows) return;
    for (int c = 0; c < f; ++c) {
        float mu  = stats[c] * invcount;
        float var = stats[f + c] * invcount - mu * mu;
        float v = g[c] * (z[row * (long)f + c] - mu) * rsqrtf(var + 1e-5f) + be[c];
        v = fmaxf(v, 0.0f);
        if (res) v += res[row * (long)f + c];
        out[row * (long)f + c] = v;
    }
}

// h[:, :, 0] = temporal_features (layout h: [N, T, ldh])
__global__ void k_inith(const float* __restrict__ tf, float* __restrict__ h,
                        long NT, int ldh)
{
    long i = (long)blockIdx.x * blockDim.x + threadIdx.x;
    if (i < NT) h[i * (long)ldh] = tf[i];
}

// conv1d over T, kernel 3, pad 1: h[node, t, coff+o] = bc[o] + sum_{i,k} Wc[o,i,k]*x[node,t+k-1,i]
__global__ void k_conv1d(const float* __restrict__ x, const float* __restrict__ Wc,
                         const float* __restrict__ bc, float* __restrict__ h,
                         long Nn, int T, int f, int ldh, int coff)
{
    long i = (long)blockIdx.x * blockDim.x + threadIdx.x;
    long total = Nn * (long)f * T;
    if (i >= total) return;
    int  t = (int)(i % T);
    long r = i / T;
    int  o = (int)(r % f);
    long node = r / f;
    float acc = bc[o];
    for (int k = 0; k < 3; ++k) {
        int tt = t + k - 1;
        if (tt < 0 || tt >= T) continue;
        const float* xp = x + (node * T + tt) * (long)f;
        for (int ii = 0; ii < f; ++ii) acc += Wc[(o * f + ii) * 3 + k] * xp[ii];
    }
    h[(node * T + t) * (long)ldh + coff + o] = acc;
}

__global__ void k_extract(const float* __restrict__ s, float* __restrict__ ssel,
                          int B, int P, int f)
{
    int i = blockIdx.x * blockDim.x + threadIdx.x;
    if (i >= B * f) return;
    int b = i / f, c = i % f;
    ssel[i] = s[(long)b * P * f + c];
}

// ext FC (8->8 relu ->10 relu ->10) + readout: sigmoid(relu([s,t,e]) @ W + b)
__global__ void k_final(const float* __restrict__ ext,
                        const float* __restrict__ eW0, const float* __restrict__ eb0,
                        const float* __restrict__ eW1, const float* __restrict__ eb1,
                        const float* __restrict__ eW2, const float* __restrict__ eb2,
                        const float* __restrict__ oW,  const float* __restrict__ ob,
                        const float* __restrict__ ssel, const float* __restrict__ h,
                        int B, int P, int T, int ldh, float* __restrict__ out)
{
    int b = blockIdx.x * blockDim.x + threadIdx.x;
    if (b >= B) return;
    float l1[8], l2[10], l3[10];
    for (int o = 0; o < 8; ++o) {
        float s = eb0[o];
        for (int i = 0; i < 8; ++i) s += ext[b * 8 + i] * eW0[i * 8 + o];
        l1[o] = fmaxf(s, 0.0f);
    }
    for (int o = 0; o < 10; ++o) {
        float s = eb1[o];
        for (int i = 0; i < 8; ++i) s += l1[i] * eW1[i * 10 + o];
        l2[o] = fmaxf(s, 0.0f);
    }
    for (int o = 0; o < 10; ++o) {
        float s = eb2[o];
        for (int i = 0; i < 10; ++i) s += l2[i] * eW2[i * 10 + o];
        l3[o] = s;                                 // no relu on last fc layer
    }
    const long node = (long)b * P;
    float acc = ob[0];
    for (int c = 0; c < 10; ++c)                   // s[idx]
        acc += fmaxf(ssel[b * 10 + c], 0.0f) * oW[c];
    for (int c = 0; c < 20; ++c) {                 // t[idx] = mean over T
        float m = 0.0f;
        for (int t = 0; t < T; ++t) m += h[(node * T + t) * (long)ldh + c];
        acc += fmaxf(m / (float)T, 0.0f) * oW[10 + c];
    }
    for (int c = 0; c < 10; ++c)                   // e
        acc += fmaxf(l3[c], 0.0f) * oW[30 + c];
    out[b] = 1.0f / (1.0f + expf(-acc));
}

// ---------------------------------------------------------------------------
static inline dim3 nb(long n, int bs = 256) { return dim3((unsigned)((n + bs - 1) / bs)); }

// Dispatch runtime (fi,fo) to the compile-time WMMA kernel.
static void launch_dense(hipStream_t stream, const float* in, int ldi, int fi,
                         const float* W, const float* bias, const float* scale,
                         int rdiv, float* out, int ldo, int fo, long rows, int relu)
{
    const long pairs = (rows + 31) / 32;           // 2 row tiles per wave
    const dim3 grid((unsigned)((pairs + 3) / 4));  // 4 waves per block
    const dim3 block(128);
#define DCASE(FI_, FO_)                                                         \
    if (fi == FI_ && fo == FO_) {                                               \
        k_wmma_dense_t<FI_, FO_><<<grid, block, 0, stream>>>(                   \
            in, ldi, W, bias, scale, rdiv, out, ldo, rows, relu);               \
        return;                                                                 \
    }
    DCASE(22, 20) DCASE(20, 15) DCASE(15, 15) DCASE(15, 14) DCASE(14, 13)
    DCASE(13, 12) DCASE(12, 11) DCASE(11, 10) DCASE(1, 1)   DCASE(1, 4)
    DCASE(4, 4)   DCASE(5, 5)   DCASE(10, 10)
#undef DCASE
    k_dense_fallback<<<nb(rows), 256, 0, stream>>>(in, ldi, fi, W, bias, scale,
                                                   rdiv, out, ldo, fo, rows, relu);
}

extern "C" void kernel_launch(void* const* d_in, const int* in_sizes, int n_in,
                              void* d_out, int out_size, void* d_ws, size_t ws_size,
                              hipStream_t stream)
{
    // ---- input layout (top-level in setup_inputs() insertion order; params as
    //      a JAX pytree: sorted dict keys -> ext_fc, out, sgcn, spatial_fc, st;
    //      per-layer dicts W,b,be,g; st blocks Wc,bc,gcn). 111 inputs total. ----
    if (n_in < 111) return;
    const float* spatial  = (const float*)d_in[0];   // [N,22]
    const float* temporal = (const float*)d_in[1];   // [N,1,24]
    const float* ext      = (const float*)d_in[2];   // [B,8]
    const int*   src      = (const int*)d_in[n_in - 2];
    const int*   dst      = (const int*)d_in[n_in - 1];

    const long N  = in_sizes[0] / 22;
    const int  T  = (int)(in_sizes[1] / N);
    const int  B  = in_sizes[2] / 8;
    const int  P  = (int)(N / B);
    const long E  = in_sizes[n_in - 1];
    const long NT = N * T;

    auto p = [&](int i) -> const float* { return (const float*)d_in[i]; };

    // ---- workspace carve-up (~316 MB) ----
    float* w     = (float*)d_ws;
    float* on    = w;                 // N   : outdeg^-1/2
    float* inn   = w + N;             // N   : indeg^-1/2
    int*   deg   = (int*)(w + 2 * N); // N   : int scratch
    float* stats = w + 3 * N;         // 64  : per-layer BN sum/sumsq
    float* ssel  = w + 3 * N + 64;    // B*10
    float* big   = w + 3 * N + 512;
    // spatial view of big region
    float* SA = big;                  // N*20
    float* SB = big + N * 20;
    float* SC = big + 2 * N * 20;
    float* S  = big + 3 * N * 20;
    // temporal view of big region (reused after spatial branch finishes)
    float* H  = big;                  // NT*20 : h in [N, T, 20] layout
    float* X0 = big + NT * 20;        // NT*10 : AGG buffer
    float* X1 = big + NT * 30;        // NT*10 : layer activations
    float* Yb = big + NT * 40;        // NT*10 : matmul out / z
    {
        size_t need = (size_t)(3 * N + 512 + NT * 50) * sizeof(float);
        if (ws_size < need) return;   // workspace too small — bail safely
    }

    // ---- degree normalizers ----
    k_zero<<<nb(N), 256, 0, stream>>>((float*)deg, N);
    k_count<<<nb(E), 256, 0, stream>>>(src, deg, E);
    k_rsqrtdeg<<<nb(N), 256, 0, stream>>>(deg, on, N);
    k_zero<<<nb(N), 256, 0, stream>>>((float*)deg, N);
    k_count<<<nb(E), 256, 0, stream>>>(dst, deg, E);
    k_rsqrtdeg<<<nb(N), 256, 0, stream>>>(deg, inn, N);

    // one graph-conv layer: out = relu(BN( agg(on.*in@W)[edges] .* inn + b )) [+res]
    auto gconv = [&](const float* in, int ldi, int fi, int fo,
                     const float* W, const float* bias, const float* g, const float* be,
                     float* Y, float* AGG, float* outp, const float* res,
                     long rows, int rdiv) {
        launch_dense(stream, in, ldi, fi, W, nullptr, on, rdiv, Y, fo, fo, rows, 0);
        k_zero<<<nb(rows * fo), 256, 0, stream>>>(AGG, rows * (long)fo);
        k_zero<<<1, 64, 0, stream>>>(stats, 2 * fo);
        const int vpn = rdiv * fo;                      // values per node
        if ((vpn & 3) == 0)
            k_scatter4<<<nb(E * (long)(vpn >> 2)), 256, 0, stream>>>(src, dst, Y, AGG, E, vpn);
        else
            k_scatter<<<nb(E * (long)vpn), 256, 0, stream>>>(src, dst, Y, AGG, E, vpn);
        k_finish<<<nb(rows), 256, 0, stream>>>(AGG, inn, rdiv, bias, Y, stats, rows, fo);
        k_bn<<<nb(rows), 256, 0, stream>>>(Y, stats, g, be, res, outp, rows, fo,
                                           1.0f / (float)rows);
    };

    // ================= spatial branch =================
    launch_dense(stream, spatial, 22, 22, p(63), p(64), nullptr, 1, SA, 20, 20, N, 1);
    launch_dense(stream, SA, 20, 20, p(65), p(66), nullptr, 1, S, 15, 15, N, 0);

    for (int gidx = 0; gidx < 2; ++gidx) {              // two residual 4-layer GCNs
        int base = 11 + gidx * 16;                      // leaves: W,b,be,g per layer
        gconv(S,  15, 15, 15, p(base + 0),  p(base + 1),  p(base + 3),  p(base + 2),  SA, SB, SC, nullptr, N, 1);
        gconv(SC, 15, 15, 15, p(base + 4),  p(base + 5),  p(base + 7),  p(base + 6),  SA, SB, SC, nullptr, N, 1);
        gconv(SC, 15, 15, 15, p(base + 8),  p(base + 9),  p(base + 11), p(base + 10), SA, SB, SC, nullptr, N, 1);
        gconv(SC, 15, 15, 15, p(base + 12), p(base + 13), p(base + 15), p(base + 14), SA, SB, S,  S,       N, 1);
    }
    {                                                   // sgcn[2]: 15->14->13->12->11->10
        const int fis[6] = {15, 14, 13, 12, 11, 10};
        const float* cur = S;
        for (int l = 0; l < 5; ++l) {
            int base = 43 + l * 4;
            gconv(cur, fis[l], fis[l], fis[l + 1], p(base), p(base + 1), p(base + 3),
                  p(base + 2), SA, SB, SC, nullptr, N, 1);
            cur = SC;
        }
    }
    k_extract<<<nb(B * 10), 256, 0, stream>>>(SC, ssel, B, P, 10); // save 32 rows

    // ================= temporal branch =================
    k_inith<<<nb(NT), 256, 0, stream>>>(temporal, H, NT, 20);
    const int bdims[3][4] = {{1, 1, 4, 4}, {5, 5, 5, 5}, {10, 10, 10, 10}};
    const int coff[3] = {1, 5, 10};
    for (int k = 0; k < 3; ++k) {
        const int base = 67 + k * 14;                   // Wc, bc, then gcn layers
        const float* Wc = p(base);
        const float* bc = p(base + 1);
        // layer 1 reads h (stride 20, first fi channels); layers 2-3 ping through X1
        for (int l = 0; l < 3; ++l) {
            const int gl = base + 2 + l * 4;
            const float* in = (l == 0) ? H : X1;
            const int ldi   = (l == 0) ? 20 : bdims[k][l];
            gconv(in, ldi, bdims[k][l], bdims[k][l + 1], p(gl), p(gl + 1), p(gl + 3),
                  p(gl + 2), Yb, X0, X1, nullptr, NT, T);
        }
        const int fo = bdims[k][3];
        k_conv1d<<<nb(N * (long)fo * T), 256, 0, stream>>>(X1, Wc, bc, H, N, T, fo,
                                                           20, coff[k]);
    }

    // ================= ext FC + readout =================
    k_final<<<1, 64, 0, stream>>>(ext, p(3), p(4), p(5), p(6), p(7), p(8), p(9), p(10),
                                  ssel, H, B, P, T, 20, (float*)d_out);
    (void)out_size;
}